// YOWOF_45414984188269
// MI455X (gfx1250) — compile-verified
//
#include <hip/hip_runtime.h>
#include <hip/hip_bf16.h>
#include <math.h>

// ---------------------------------------------------------------------------
// YOWOF-style detection post-processing for MI455X (gfx1250, wave32).
// Pipeline:
//   K1 score_hist : sigmoid(max logit) + argmax per anchor, fused radix hist
//   K2 thresh     : scan 4096-bin histogram for top-1000 threshold bin
//   K3 collect    : gather candidates >= threshold bin
//   K4 sort       : single-WG bitonic sort (2048, LDS) -> top 1000 desc
//   K5 decode     : gather + box decode + clip, write bboxes/scores/labels
//   K6 iou_sup    : 16x16 tiles; area_i+area_j outer-sum via
//                   V_WMMA_F32_16X16X4_F32 (rank-2 matmul), inter via VALU,
//                   packed suppression bitmask
//   K7 nms        : single wave32 sequential greedy NMS, ballot reduction
// ---------------------------------------------------------------------------

typedef __attribute__((ext_vector_type(2))) float v2f;
typedef __attribute__((ext_vector_type(8))) float v8f;

#define TOPK        1000
#define NUM_CLASSES 80
#define CONF_THRESH 0.05f
#define NMS_THRESH  0.6f
#define CTR_CLAMP   32.0f
#define SCALE_CLAMP 4.1351665567423560f   // log(1000/16)
#define IMG_SIZE_F  2048.0f
#define NBINS       4096
#define CAND_CAP    2048

__device__ __forceinline__ unsigned order_key(float s) {
    unsigned u = __float_as_uint(s);
    return u ^ ((u >> 31) ? 0xFFFFFFFFu : 0x80000000u);
}

// --- K1: per-anchor max/argmax of logits, sigmoid score, fused histogram ----
__global__ void score_hist_kernel(const float* __restrict__ cls,
                                  float* __restrict__ scores,
                                  int* __restrict__ labels,
                                  unsigned* __restrict__ hist, int M) {
    int m = blockIdx.x * blockDim.x + threadIdx.x;
    if (m >= M) return;
    const float4* row = (const float4*)(cls + (size_t)m * NUM_CLASSES);
    __builtin_prefetch(row + 20, 0, 1);   // global_prefetch_b8: next row
    float best = -3.4e38f; int bi = 0;
#pragma unroll
    for (int q = 0; q < NUM_CLASSES / 4; ++q) {
        float4 v = row[q];
        int c = q * 4;
        if (v.x > best) { best = v.x; bi = c;     }
        if (v.y > best) { best = v.y; bi = c + 1; }
        if (v.z > best) { best = v.z; bi = c + 2; }
        if (v.w > best) { best = v.w; bi = c + 3; }
    }
    float s = 1.0f / (1.0f + __expf(-best));   // sigmoid monotone => max ok
    scores[m] = s;
    labels[m] = bi;
    atomicAdd(&hist[order_key(s) >> 20], 1u);
}

// --- K2: find threshold bin so that count(bins >= t) >= TOPK ----------------
__global__ void thresh_kernel(const unsigned* __restrict__ hist,
                              unsigned* __restrict__ threshBin) {
    if (threadIdx.x != 0 || blockIdx.x != 0) return;
    unsigned cum = 0, t = 0;
    for (int b = NBINS - 1; b >= 0; --b) {
        cum += hist[b];
        if (cum >= TOPK) { t = (unsigned)b; break; }
    }
    *threshBin = t;
}

// --- K3: collect candidate (score, idx) above threshold bin -----------------
__global__ void collect_kernel(const float* __restrict__ scores,
                               const unsigned* __restrict__ threshBin,
                               unsigned* __restrict__ counter,
                               float* __restrict__ candScore,
                               int* __restrict__ candIdx, int M) {
    int m = blockIdx.x * blockDim.x + threadIdx.x;
    if (m >= M) return;
    float s = scores[m];
    if ((order_key(s) >> 20) >= *threshBin) {
        unsigned pos = atomicAdd(counter, 1u);
        if (pos < CAND_CAP) { candScore[pos] = s; candIdx[pos] = m; }
    }
}

// --- K4: single-workgroup bitonic sort (descending), emit top 1024 ----------
__global__ void __launch_bounds__(1024)
sort_kernel(const float* __restrict__ candScore, const int* __restrict__ candIdx,
            const unsigned* __restrict__ counter,
            float* __restrict__ topScore, int* __restrict__ topIdx) {
    __shared__ float s[CAND_CAP];
    __shared__ int   id[CAND_CAP];
    unsigned tid = threadIdx.x;
    unsigned n = *counter; if (n > CAND_CAP) n = CAND_CAP;
    for (unsigned e = tid; e < CAND_CAP; e += 1024) {
        if (e < n) { s[e] = candScore[e]; id[e] = candIdx[e]; }
        else       { s[e] = -1.0f;        id[e] = -1;         }
    }
    __syncthreads();
    for (unsigned k = 2; k <= CAND_CAP; k <<= 1) {
        for (unsigned j = k >> 1; j > 0; j >>= 1) {
            for (unsigned t = tid; t < CAND_CAP; t += 1024) {
                unsigned ixj = t ^ j;
                if (ixj > t) {
                    bool up = ((t & k) == 0);
                    bool sw = up ? (s[t] < s[ixj]) : (s[t] > s[ixj]);
                    if (sw) {
                        float ts = s[t]; s[t] = s[ixj]; s[ixj] = ts;
                        int   ti = id[t]; id[t] = id[ixj]; id[ixj] = ti;
                    }
                }
            }
            __syncthreads();
        }
    }
    if (tid < 1024) { topScore[tid] = s[tid]; topIdx[tid] = id[tid]; }
}

// --- K5: gather + decode + clip; write bboxes/scores/labels + NMS inputs ----
__global__ void decode_kernel(const float* __restrict__ reg,
                              const float* __restrict__ anch,
                              const int* __restrict__ labels_all,
                              const float* __restrict__ topScore,
                              const int* __restrict__ topIdx,
                              float* __restrict__ out,
                              float* __restrict__ topBox,
                              int* __restrict__ topLab,
                              float* __restrict__ topArea,
                              int* __restrict__ validArr) {
    int t = blockIdx.x * blockDim.x + threadIdx.x;
    if (t >= TOPK) return;
    int idx = topIdx[t];
    float sc = topScore[t];
    float bx0 = 0.f, by0 = 0.f, bx1 = 0.f, by1 = 0.f;
    int lab = -1;
    if (idx >= 0) {
        lab = labels_all[idx];
        float4 r = ((const float4*)reg)[idx];
        float4 a = ((const float4*)anch)[idx];
        float ox = fminf(fmaxf(r.x * a.z, -CTR_CLAMP), CTR_CLAMP);
        float oy = fminf(fmaxf(r.y * a.w, -CTR_CLAMP), CTR_CLAMP);
        float cx = a.x + ox, cy = a.y + oy;
        float w = a.z * __expf(fminf(r.z, SCALE_CLAMP));
        float h = a.w * __expf(fminf(r.w, SCALE_CLAMP));
        bx0 = fminf(fmaxf((cx - 0.5f * w) / IMG_SIZE_F, 0.f), 1.f);
        by0 = fminf(fmaxf((cy - 0.5f * h) / IMG_SIZE_F, 0.f), 1.f);
        bx1 = fminf(fmaxf((cx + 0.5f * w) / IMG_SIZE_F, 0.f), 1.f);
        by1 = fminf(fmaxf((cy + 0.5f * h) / IMG_SIZE_F, 0.f), 1.f);
    } else {
        sc = -1.0f;
    }
    out[t * 4 + 0] = bx0; out[t * 4 + 1] = by0;
    out[t * 4 + 2] = bx1; out[t * 4 + 3] = by1;
    out[4 * TOPK + t] = sc;
    out[5 * TOPK + t] = (float)lab;
    topBox[t * 4 + 0] = bx0; topBox[t * 4 + 1] = by0;
    topBox[t * 4 + 2] = bx1; topBox[t * 4 + 3] = by1;
    topLab[t]  = lab;
    topArea[t] = (bx1 - bx0) * (by1 - by0);
    validArr[t] = (sc >= CONF_THRESH) ? 1 : 0;
}

// --- K6: 16x16 IoU tiles; area outer-sum via V_WMMA_F32_16X16X4_F32 ---------
// A(16x4) rows = [area_i, 1, 0, 0]; B(4x16) cols = [1, area_j, 0, 0]^T
// => D[i][j] = area_i + area_j, accumulated by the matrix pipe.
__global__ void __launch_bounds__(32)
iou_sup_kernel(const float* __restrict__ topBox,
               const int* __restrict__ topLab,
               const float* __restrict__ topArea,
               unsigned* __restrict__ sup) {
    __shared__ float rB[16][4], cB[16][4];
    __shared__ float rA[16], cA[16];
    __shared__ int   rL[16], cL[16];

    int lane = threadIdx.x;
    int half = lane >> 4;      // 0: lanes 0-15, 1: lanes 16-31
    int l    = lane & 15;
    int ti = blockIdx.y, tj = blockIdx.x;

    if (half == 0) {
        int gi = ti * 16 + l;
        bool ok = gi < TOPK;
        float4 b = ok ? ((const float4*)topBox)[gi] : make_float4(0.f, 0.f, 0.f, 0.f);
        rB[l][0] = b.x; rB[l][1] = b.y; rB[l][2] = b.z; rB[l][3] = b.w;
        rA[l] = ok ? topArea[gi] : 0.f;
        rL[l] = ok ? topLab[gi] : -1;
    } else {
        int gj = tj * 16 + l;
        bool ok = gj < TOPK;
        float4 b = ok ? ((const float4*)topBox)[gj] : make_float4(0.f, 0.f, 0.f, 0.f);
        cB[l][0] = b.x; cB[l][1] = b.y; cB[l][2] = b.z; cB[l][3] = b.w;
        cA[l] = ok ? topArea[gj] : 0.f;
        cL[l] = ok ? topLab[gj] : -2;
    }
    __syncthreads();

    // A 16x4 f32: lanes 0-15 hold K=0 (VGPR0) / K=1 (VGPR1); lanes 16-31 K=2/3.
    v2f a, b;
    a.x = (half == 0) ? rA[l] : 0.f;   // K0 = area_i   (K2 = 0)
    a.y = (half == 0) ? 1.0f  : 0.f;   // K1 = 1        (K3 = 0)
    b.x = (half == 0) ? 1.0f  : 0.f;   // K0 row = 1
    b.y = (half == 0) ? cA[l] : 0.f;   // K1 row = area_j
    v8f c = {0.f, 0.f, 0.f, 0.f, 0.f, 0.f, 0.f, 0.f};
    v8f d = __builtin_amdgcn_wmma_f32_16x16x4_f32(
        /*neg_a=*/false, a, /*neg_b=*/false, b,
        /*c_mod=*/(short)0, c, /*reuse_a=*/false, /*reuse_b=*/false);

#pragma unroll
    for (int r = 0; r < 8; ++r) {
        int Mi = r + 8 * half;         // C/D layout: lane<16 -> M=r, else M=r+8
        int Nj = l;
        float xx1 = fmaxf(rB[Mi][0], cB[Nj][0]);
        float yy1 = fmaxf(rB[Mi][1], cB[Nj][1]);
        float xx2 = fminf(rB[Mi][2], cB[Nj][2]);
        float yy2 = fminf(rB[Mi][3], cB[Nj][3]);
        float w = fmaxf(1e-10f, xx2 - xx1);
        float h = fmaxf(1e-10f, yy2 - yy1);
        float inter = w * h;
        float iou = inter / (d[r] - inter + 1e-10f);  // d[r] = area_i + area_j
        bool bit = (iou > NMS_THRESH) && (rL[Mi] == cL[Nj]);
        int gi = ti * 16 + Mi, gj = tj * 16 + Nj;
        if (bit && gi < TOPK && gj < TOPK)
            atomicOr(&sup[gi * 32 + (gj >> 5)], 1u << (gj & 31));
    }
}

// --- K7: sequential greedy NMS, one wave32; keep-mask in registers ----------
__global__ void __launch_bounds__(32)
nms_kernel(const unsigned* __restrict__ sup,
           const int* __restrict__ validArr,
           float* __restrict__ out) {
    int lane = threadIdx.x;
    unsigned keepw = 0;                 // lane t owns keep bits [32t, 32t+31]
    for (int i = 0; i < TOPK; ++i) {
        unsigned supw = sup[i * 32 + lane];
        int wi = i >> 5, bi = i & 31;
        unsigned mask = (lane < wi) ? 0xFFFFFFFFu
                      : (lane == wi) ? ((1u << bi) - 1u) : 0u;
        bool ov = (keepw & supw & mask) != 0u;
        unsigned ball = __builtin_amdgcn_ballot_w32(ov);
        bool kp = (validArr[i] != 0) && (ball == 0u);
        if (kp && lane == wi) keepw |= (1u << bi);
        if (lane == 0) out[6 * TOPK + i] = kp ? 1.0f : 0.0f;
    }
}

// ---------------------------------------------------------------------------
extern "C" void kernel_launch(void* const* d_in, const int* in_sizes, int n_in,
                              void* d_out, int out_size, void* d_ws, size_t ws_size,
                              hipStream_t stream) {
    const float* cls  = (const float*)d_in[0];
    const float* reg  = (const float*)d_in[1];
    const float* anch = (const float*)d_in[2];
    float* out = (float*)d_out;
    const int M = in_sizes[0] / NUM_CLASSES;

    char* ws = (char*)d_ws;
    size_t off = 0;
    float*    scores_all = (float*)(ws + off);    off += (size_t)M * 4;
    int*      labels_all = (int*)(ws + off);      off += (size_t)M * 4;
    unsigned* hist       = (unsigned*)(ws + off); off += NBINS * 4;
    unsigned* counter    = (unsigned*)(ws + off); off += 16;
    unsigned* threshBin  = (unsigned*)(ws + off); off += 16;
    float*    candScore  = (float*)(ws + off);    off += CAND_CAP * 4;
    int*      candIdx    = (int*)(ws + off);      off += CAND_CAP * 4;
    float*    topScore   = (float*)(ws + off);    off += 1024 * 4;
    int*      topIdx     = (int*)(ws + off);      off += 1024 * 4;
    float*    topBox     = (float*)(ws + off);    off += 1024 * 4 * 4;
    int*      topLab     = (int*)(ws + off);      off += 1024 * 4;
    float*    topArea    = (float*)(ws + off);    off += 1024 * 4;
    int*      validArr   = (int*)(ws + off);      off += 1024 * 4;
    unsigned* sup        = (unsigned*)(ws + off); off += 1024 * 32 * 4;
    (void)ws_size; (void)n_in; (void)out_size;

    hipMemsetAsync(hist, 0, NBINS * 4, stream);
    hipMemsetAsync(counter, 0, 4, stream);
    hipMemsetAsync(sup, 0, 1024 * 32 * 4, stream);

    int blocks = (M + 255) / 256;
    score_hist_kernel<<<blocks, 256, 0, stream>>>(cls, scores_all, labels_all, hist, M);
    thresh_kernel<<<1, 1, 0, stream>>>(hist, threshBin);
    collect_kernel<<<blocks, 256, 0, stream>>>(scores_all, threshBin, counter,
                                               candScore, candIdx, M);
    sort_kernel<<<1, 1024, 0, stream>>>(candScore, candIdx, counter, topScore, topIdx);
    decode_kernel<<<4, 256, 0, stream>>>(reg, anch, labels_all, topScore, topIdx,
                                         out, topBox, topLab, topArea, validArr);
    dim3 iouGrid(63, 63);
    iou_sup_kernel<<<iouGrid, 32, 0, stream>>>(topBox, topLab, topArea, sup);
    nms_kernel<<<1, 32, 0, stream>>>(sup, validArr, out);
}